// TreeVRNN_61366492725367
// MI455X (gfx1250) — compile-verified
//
#include <hip/hip_runtime.h>

// B=32, L=64, D=512, K=200
#define BB 32
#define LL 64
#define DD 512
#define KK 200
#define NT 13  // ceil(200/16) tiles along K

typedef float v2f __attribute__((ext_vector_type(2)));
typedef float v8f __attribute__((ext_vector_type(8)));

// Fast tanh from CDNA5 trans ops: tanh(x) = 1 - 2/(exp2(2x*log2e)+1)
// Saturates correctly to +/-1 on overflow (exp2->inf -> rcp->0).
__device__ __forceinline__ float fast_tanh(float x) {
    float e = __builtin_amdgcn_exp2f(x * 2.885390081777927f); // 2*log2(e)
    return 1.0f - 2.0f * __builtin_amdgcn_rcpf(e + 1.0f);
}

// One wave per 16x16 output tile of  out[m, n] = sum_d sent[m, d] * W[n, d]
// m in [0,2048) (= b*64 + i), n in [0,200) (= k channel).
// Result stored TRANSPOSED as outT[b][k][i]  (i contiguous) for stage 2.
__global__ __launch_bounds__(32) void proj_gemm_wmma(
    const float* __restrict__ sent,  // [2048, 512] row-major
    const float* __restrict__ W,     // [200, 512] row-major
    float* __restrict__ outT)        // [32][200][64]
{
    const int mt   = blockIdx.x;          // 0..127
    const int nt   = blockIdx.y;          // 0..12
    const int lane = threadIdx.x;         // 0..31
    const int lo   = lane & 15;
    const int hi   = lane >> 4;           // 0 or 1

    const int m0 = mt * 16;
    const int n0 = nt * 16;

    // A-frag: lane holds M = m0+lo, K = d0 + 2*hi + {0,1}  -> float2 load
    const float* pa = sent + (size_t)(m0 + lo) * DD + 2 * hi;
    // B-frag: lane holds N = n0+lo, K = d0 + 2*hi + {0,1}  (W is [n][d])
    int nrow = n0 + lo;
    if (nrow > KK - 1) nrow = KK - 1;     // clamp tail tile (stores masked below)
    const float* pw = W + (size_t)nrow * DD + 2 * hi;

    v8f c0 = {};
    v8f c1 = {};
    // D = 512, k-step 4 => 128 WMMAs; two accumulators break the RAW chain.
#pragma unroll 4
    for (int d = 0; d < DD; d += 8) {
        v2f a0 = *(const v2f*)(pa + d);
        v2f b0 = *(const v2f*)(pw + d);
        c0 = __builtin_amdgcn_wmma_f32_16x16x4_f32(false, a0, false, b0,
                                                   (short)0, c0, false, false);
        v2f a1 = *(const v2f*)(pa + d + 4);
        v2f b1 = *(const v2f*)(pw + d + 4);
        c1 = __builtin_amdgcn_wmma_f32_16x16x4_f32(false, a1, false, b1,
                                                   (short)0, c1, false, false);
    }
    v8f c = c0 + c1;

    // C layout: VGPR r, lanes0-15 -> M=m0+r, lanes16-31 -> M=m0+r+8; N=n0+lo.
    const int kch = n0 + lo;
    if (kch < KK) {
        const int b      = m0 / LL;       // 16 | 64 so tile sits in one batch
        const int i_base = (m0 % LL) + 8 * hi;
        float* po = outT + ((size_t)b * KK + kch) * LL;
#pragma unroll
        for (int r = 0; r < 8; ++r) {
            po[i_base + r] = c[r];
        }
    }
}

// One 64-thread block per (b, i); lane = j.
// out[b,i,j] = (i<=j<=tgt[b]) ? tanh( sum_k tanh(A[b,i,k]+Bm[b,j,k]+bias[k])*s[k] ) : 0
__global__ __launch_bounds__(64) void tree_potentials(
    const float* __restrict__ At,    // [32][200][64]  (A transposed, i contiguous)
    const float* __restrict__ BmT,   // [32][200][64]  (Bm transposed, j contiguous)
    const float* __restrict__ bias,  // [200]
    const float* __restrict__ s,     // [200]
    const int*   __restrict__ tgt,   // [32]
    float* __restrict__ out)         // [32][64][64]
{
    const int b = blockIdx.x >> 6;
    const int i = blockIdx.x & 63;
    const int j = threadIdx.x;

    float* po = out + ((size_t)b * LL + i) * LL + j;   // coalesced across lanes

    const int t = tgt[b];
    if (i > t) {            // whole row masked: j >= i > tgt
        *po = 0.0f;
        return;
    }

    __shared__ float smA[KK];   // A[b, i, k] + bias[k]  (uniform over lanes)
    __shared__ float smS[KK];

    const float* pa = At + (size_t)b * KK * LL + i;
    for (int k = j; k < KK; k += 64) {
        smA[k] = pa[(size_t)k * LL] + bias[k];
        smS[k] = s[k];
    }
    __syncthreads();

    const float* pb = BmT + (size_t)b * KK * LL + j;   // lane-contiguous per k
    float acc = 0.0f;
#pragma unroll 4
    for (int k = 0; k < KK; ++k) {
        float x = pb[(size_t)k * LL] + smA[k];
        acc = fmaf(fast_tanh(x), smS[k], acc);
    }

    float r = fast_tanh(acc);
    *po = (j >= i && j <= t) ? r : 0.0f;
}

extern "C" void kernel_launch(void* const* d_in, const int* in_sizes, int n_in,
                              void* d_out, int out_size, void* d_ws, size_t ws_size,
                              hipStream_t stream) {
    const float* sent = (const float*)d_in[0];   // [32,64,512]
    const float* W1   = (const float*)d_in[1];   // [200,512]
    const float* W2   = (const float*)d_in[2];   // [200,512]
    const float* bias = (const float*)d_in[3];   // [200]
    const float* s    = (const float*)d_in[4];   // [200]
    const int*   tgt  = (const int*)d_in[5];     // [32]
    float* out = (float*)d_out;                  // [32,64,64]

    float* At  = (float*)d_ws;                   // [32][200][64]
    float* BmT = At + (size_t)BB * KK * LL;      // [32][200][64]

    dim3 ggrid((BB * LL) / 16, NT);
    proj_gemm_wmma<<<ggrid, 32, 0, stream>>>(sent, W1, At);
    proj_gemm_wmma<<<ggrid, 32, 0, stream>>>(sent, W2, BmT);

    tree_potentials<<<BB * LL, 64, 0, stream>>>(At, BmT, bias, s, tgt, out);
}